// GNN_5016521802376
// MI455X (gfx1250) — compile-verified
//
#include <hip/hip_runtime.h>
#include <hip/hip_bf16.h>

typedef __attribute__((ext_vector_type(2))) float v2f;
typedef __attribute__((ext_vector_type(8))) float v8f;

#define DD 64   // hidden dim D == attn dim A == 64

// ---------------------------------------------------------------------------
// Zero the segment-sum buffer.
__global__ void gnn_zero_kernel(float* __restrict__ p, long n) {
    long i = (long)blockIdx.x * blockDim.x + threadIdx.x;
    if (i < n) p[i] = 0.0f;
}

// ---------------------------------------------------------------------------
// wr_pre[r][a] = dot(rela_embed[r], Wr_attn[a])   (tiny: 401x64)
__global__ void gnn_relpre_kernel(const float* __restrict__ rela,
                                  const float* __restrict__ Wr,
                                  float* __restrict__ out, int RV) {
    int t = blockIdx.x * blockDim.x + threadIdx.x;
    if (t >= RV * DD) return;
    int r = t / DD, a = t % DD;
    const float* x = rela + (long)r * DD;
    const float* w = Wr   + (long)a * DD;
    float acc = 0.0f;
    #pragma unroll
    for (int d = 0; d < DD; ++d) acc = fmaf(x[d], w[d], acc);
    out[t] = acc;
}

// ---------------------------------------------------------------------------
// qr_pre[b][a] = dot(rela_embed[q_rel[b]], Wqr_attn[a]) + Wqr_b[a]  (1024x64)
__global__ void gnn_qrpre_kernel(const int* __restrict__ q_rel,
                                 const float* __restrict__ rela,
                                 const float* __restrict__ Wqr,
                                 const float* __restrict__ Wqr_b,
                                 float* __restrict__ out, int B) {
    int t = blockIdx.x * blockDim.x + threadIdx.x;
    if (t >= B * DD) return;
    int b = t / DD, a = t % DD;
    const float* x = rela + (long)q_rel[b] * DD;
    const float* w = Wqr  + (long)a * DD;
    float acc = Wqr_b[a];
    #pragma unroll
    for (int d = 0; d < DD; ++d) acc = fmaf(x[d], w[d], acc);
    out[t] = acc;
}

// ---------------------------------------------------------------------------
// C[n][j] = sum_d A[n][d] * W[j][d]   with A: N x 64 row-major, W: 64 x 64
// row-major (used transposed).  fp32 WMMA, 16x16 output tile, K=4 per issue.
//
// ISA 7.12.2 layouts (wave32):
//   A 16x4 f32 : lane = (k>=2)*16 + m,  vgpr = k&1  ->  K = vgpr + 2*(lane>=16)
//   B 4x16 f32 : vgpr r holds K = r + 2*(lane>=16), N = lane&15
//   C 16x16 f32: vgpr r holds M = r + 8*(lane>=16), N = lane&15
__global__ void gnn_gemm_nt64_wmma(const float* __restrict__ A,
                                   const float* __restrict__ W,
                                   float* __restrict__ C, int N) {
    int tid   = blockIdx.x * blockDim.x + threadIdx.x;
    int wave  = tid >> 5;
    int lane  = tid & 31;
    int ntile = (N + 15) >> 4;
    int nt = wave >> 2;          // which 16-row block of A / C
    int jt = wave & 3;           // which 16-col block of C (4 blocks of 16 = 64)
    if (nt >= ntile) return;     // wave-uniform exit: EXEC stays all-ones

    int m     = lane & 15;
    int khalf = lane >> 4;
    int arow  = nt * 16 + m;
    if (arow >= N) arow = N - 1;        // clamp loads; stores are guarded
    int wrow  = jt * 16 + m;            // B column index == row of W

    const float* ap = A + (long)arow * DD;
    const float* wp = W + (long)wrow * DD;

    v8f c = {};
    #pragma unroll
    for (int k0 = 0; k0 < DD; k0 += 4) {
        v2f a, b;
        a.x = ap[k0 + 2 * khalf + 0];
        a.y = ap[k0 + 2 * khalf + 1];
        b.x = wp[k0 + 2 * khalf + 0];
        b.y = wp[k0 + 2 * khalf + 1];
        c = __builtin_amdgcn_wmma_f32_16x16x4_f32(
                /*neg_a=*/false, a, /*neg_b=*/false, b,
                /*c_mod=*/(short)0, c, /*reuse_a=*/false, /*reuse_b=*/false);
    }

    int ccol = jt * 16 + (lane & 15);
    #pragma unroll
    for (int r = 0; r < 8; ++r) {
        int crow = nt * 16 + r + 8 * khalf;
        if (crow < N) C[(long)crow * DD + ccol] = c[r];
    }
}

// ---------------------------------------------------------------------------
// Edge loop: each wave32 loops over edges_per_wave consecutive edges (each lane owns 2 of the 64 dims).
//   pre  = ws_pre[sub] + wr_pre[rel] + qr_pre[r_idx]
//   alpha= sigmoid(dot(relu(pre), w_alpha) + w_alpha_b)
//   seg[obj] += alpha * (hidden[sub] + rela_embed[rel])     (f32 atomics)
__global__ void gnn_edge_kernel(const int* __restrict__ edges,
                                const float* __restrict__ ws_pre,
                                const float* __restrict__ wr_pre,
                                const float* __restrict__ qr_pre,
                                const float* __restrict__ hidden,
                                const float* __restrict__ rela,
                                const float* __restrict__ w_alpha,
                                const float* __restrict__ w_alpha_b,
                                float* __restrict__ seg,
                                int E, int edges_per_wave) {
    int tid  = blockIdx.x * blockDim.x + threadIdx.x;
    int wave = tid >> 5;
    int lane = tid & 31;

    long e0 = (long)wave * edges_per_wave;
    long e1 = e0 + edges_per_wave;
    if (e1 > E) e1 = E;
    if (e0 >= e1) return;

    const float2 wa = reinterpret_cast<const float2*>(w_alpha)[lane];
    const float  wb = w_alpha_b[0];

    for (long e = e0; e < e1; ++e) {
        const int* ed = edges + e * 6;
        int r_idx = ed[0];
        int rel   = ed[2];
        int sub   = ed[4];
        int obj   = ed[5];

        float2 ps = reinterpret_cast<const float2*>(ws_pre + (long)sub   * DD)[lane];
        float2 pr = reinterpret_cast<const float2*>(wr_pre + (long)rel   * DD)[lane];
        float2 pq = reinterpret_cast<const float2*>(qr_pre + (long)r_idx * DD)[lane];

        float p0 = fmaxf(ps.x + pr.x + pq.x, 0.0f);
        float p1 = fmaxf(ps.y + pr.y + pq.y, 0.0f);
        float part = fmaf(p0, wa.x, p1 * wa.y);

        #pragma unroll
        for (int off = 16; off > 0; off >>= 1)
            part += __shfl_xor(part, off, 32);

        float alpha = 1.0f / (1.0f + __expf(-(part + wb)));

        float2 hs = reinterpret_cast<const float2*>(hidden + (long)sub * DD)[lane];
        float2 hr = reinterpret_cast<const float2*>(rela   + (long)rel * DD)[lane];

        float* dst = seg + (long)obj * DD + 2 * lane;
        atomicAdd(dst + 0, alpha * (hs.x + hr.x));
        atomicAdd(dst + 1, alpha * (hs.y + hr.y));
    }
}

// ---------------------------------------------------------------------------
extern "C" void kernel_launch(void* const* d_in, const int* in_sizes, int n_in,
                              void* d_out, int out_size, void* d_ws, size_t ws_size,
                              hipStream_t stream) {
    // input order per setup_inputs():
    // 0 q_sub (unused), 1 q_rel, 2 hidden, 3 edges, 4 n_node, 5 old_idx (unused),
    // 6 rela_embed, 7 Ws_attn, 8 Wr_attn, 9 Wqr_attn, 10 Wqr_b,
    // 11 w_alpha, 12 w_alpha_b, 13 W_h
    const int*   q_rel     = (const int*)  d_in[1];
    const float* hidden    = (const float*)d_in[2];
    const int*   edges     = (const int*)  d_in[3];
    const float* rela      = (const float*)d_in[6];
    const float* Ws        = (const float*)d_in[7];
    const float* Wr        = (const float*)d_in[8];
    const float* Wqr       = (const float*)d_in[9];
    const float* Wqr_b     = (const float*)d_in[10];
    const float* w_alpha   = (const float*)d_in[11];
    const float* w_alpha_b = (const float*)d_in[12];
    const float* W_h       = (const float*)d_in[13];
    float* out = (float*)d_out;

    const int B  = in_sizes[1];
    const int N  = in_sizes[2] / DD;
    const int E  = in_sizes[3] / 6;
    const int RV = in_sizes[6] / DD;

    // workspace layout (fp32): seg | ws_pre | wr_pre | qr_pre
    float* seg    = (float*)d_ws;
    float* ws_pre = seg    + (size_t)N  * DD;
    float* wr_pre = ws_pre + (size_t)N  * DD;
    float* qr_pre = wr_pre + (size_t)RV * DD;

    const int T = 256;

    // 1) zero segment sums
    long segN = (long)N * DD;
    gnn_zero_kernel<<<(int)((segN + T - 1) / T), T, 0, stream>>>(seg, segN);

    // 2) tiny relation / query-relation projections
    gnn_relpre_kernel<<<(RV * DD + T - 1) / T, T, 0, stream>>>(rela, Wr, wr_pre, RV);
    gnn_qrpre_kernel<<<(B * DD + T - 1) / T, T, 0, stream>>>(q_rel, rela, Wqr, Wqr_b,
                                                             qr_pre, B);

    // 3) node projection: ws_pre = hidden @ Ws^T   (WMMA fp32)
    {
        int ntile = (N + 15) >> 4;
        int waves = ntile * 4;
        int blocks = (waves * 32 + T - 1) / T;
        gnn_gemm_nt64_wmma<<<blocks, T, 0, stream>>>(hidden, Ws, ws_pre, N);
    }

    // 4) edge loop with atomic segment scatter
    {
        const int totalWaves = 16384;          // 2048 blocks * 8 waves
        int epw = (E + totalWaves - 1) / totalWaves;
        gnn_edge_kernel<<<totalWaves * 32 / T, T, 0, stream>>>(
            edges, ws_pre, wr_pre, qr_pre, hidden, rela,
            w_alpha, w_alpha_b, seg, E, epw);
    }

    // 5) final projection: out = seg @ W_h^T   (WMMA fp32)
    {
        int ntile = (N + 15) >> 4;
        int waves = ntile * 4;
        int blocks = (waves * 32 + T - 1) / T;
        gnn_gemm_nt64_wmma<<<blocks, T, 0, stream>>>(seg, W_h, out, N);
    }
}